// GINE_15616501088826
// MI455X (gfx1250) — compile-verified
//
#include <hip/hip_runtime.h>
#include <hip/hip_bf16.h>

#define D_FEAT 64
#define WAVE   32   // gfx1250 is wave32-only

typedef __attribute__((ext_vector_type(2))) float f32x2;
typedef __attribute__((ext_vector_type(4))) float f32x4;

// ---------------------------------------------------------------------------
// Phase 0: zero the per-node edge counters (fresh every call: no state reuse)
// ---------------------------------------------------------------------------
__global__ void k_zero_i32(int* __restrict__ p, int n) {
  int i = blockIdx.x * blockDim.x + threadIdx.x;
  const int stride = gridDim.x * blockDim.x;
  for (; i < n; i += stride) p[i] = 0;
}

// ---------------------------------------------------------------------------
// Phase 1: histogram of dst -> counts  (1.28M int atomics, native in L2)
// ---------------------------------------------------------------------------
__global__ void k_hist(const int* __restrict__ dst, int* __restrict__ counts,
                       int E) {
  int i = blockIdx.x * blockDim.x + threadIdx.x;
  const int stride = gridDim.x * blockDim.x;
  for (; i < E; i += stride)
    __hip_atomic_fetch_add(&counts[dst[i]], 1, __ATOMIC_RELAXED,
                           __HIP_MEMORY_SCOPE_AGENT);
}

// ---------------------------------------------------------------------------
// Phase 2: exclusive prefix sum over N counts in a single 1024-thread block.
// Each thread serially sums a ~98-element chunk, then an LDS Hillis-Steele
// scan combines the 1024 partials, then chunk offsets are written back.
// ---------------------------------------------------------------------------
__global__ __launch_bounds__(1024) void k_scan(const int* __restrict__ counts,
                                               int* __restrict__ offsets,
                                               int* __restrict__ cursor,
                                               int n) {
  __shared__ int part[1024];
  const int t     = threadIdx.x;
  const int chunk = (n + 1023) >> 10;
  const int begin = t * chunk;
  const int end   = min(begin + chunk, n);

  int s = 0;
  for (int i = begin; i < end; ++i) s += counts[i];
  part[t] = s;
  __syncthreads();

  for (int off = 1; off < 1024; off <<= 1) {
    int v = (t >= off) ? part[t - off] : 0;
    __syncthreads();
    part[t] += v;
    __syncthreads();
  }

  int run = (t == 0) ? 0 : part[t - 1];
  for (int i = begin; i < end; ++i) {
    offsets[i] = run;
    cursor[i]  = run;
    run += counts[i];
  }
  if (end == n) offsets[n] = run;  // all such threads write the same total
}

// ---------------------------------------------------------------------------
// Phase 3: bin edge ids by dst using the cursor array (int atomics only)
// ---------------------------------------------------------------------------
__global__ void k_fill(const int* __restrict__ dst, int* __restrict__ cursor,
                       int* __restrict__ edge_of, int E) {
  int i = blockIdx.x * blockDim.x + threadIdx.x;
  const int stride = gridDim.x * blockDim.x;
  for (; i < E; i += stride) {
    const int p = __hip_atomic_fetch_add(&cursor[dst[i]], 1, __ATOMIC_RELAXED,
                                         __HIP_MEMORY_SCOPE_AGENT);
    edge_of[p] = i;
  }
}

// ---------------------------------------------------------------------------
// Phase 4: one wave32 per node; lane l owns feature float2 [2l, 2l+1].
// Wave-cooperative index batching:
//   - lane l loads edge_of[b+l] and src[] once per 32 edges (coalesced),
//     then immediately global_prefetch's its OWN edge row (both 128B halves
//     of the 256B row) and its src node row -> up to 32 rows in flight.
//   - inner loop broadcasts (e, s) via v_readlane_b32 (wave-uniform k),
//     so the steady state is just 2 feature loads + VALU per edge.
//   - edge rows streamed ONCE -> non-temporal loads (keep node table in L2)
//   - output written once, never re-read -> non-temporal store
//   - zero atomics: deterministic per-node summation over the CSR bin
// ---------------------------------------------------------------------------
__global__ __launch_bounds__(256) void k_aggregate(
    const float* __restrict__ node, const float* __restrict__ edge,
    const int* __restrict__ src, const int* __restrict__ offsets,
    const int* __restrict__ edge_of, float* __restrict__ out, int N) {
  const int wavesPerBlock = blockDim.x / WAVE;
  const int v = blockIdx.x * wavesPerBlock + (threadIdx.x / WAVE);
  if (v >= N) return;
  const int lane = threadIdx.x & (WAVE - 1);
  const int fo   = lane * 2;

  const int beg = offsets[v];
  const int end = offsets[v + 1];

  float ax = 0.0f, ay = 0.0f;
  for (int b = beg; b < end; b += WAVE) {
    const int cnt = min(WAVE, end - b);

    int e_l = 0, s_l = 0;
    if (lane < cnt) {
      e_l = edge_of[b + lane];
      s_l = src[e_l];
      // Prefetch this lane's edge row (256B = two 128B lines) and node row.
      const float* er = edge + (size_t)e_l * D_FEAT;
      __builtin_prefetch(er, 0, 0);       // global_prefetch, line 0
      __builtin_prefetch(er + 32, 0, 0);  // global_prefetch, line 1
      const float* nr = node + (size_t)s_l * D_FEAT;
      __builtin_prefetch(nr, 0, 1);
      __builtin_prefetch(nr + 32, 0, 1);
    }

    for (int k = 0; k < cnt; ++k) {
      const int e = __builtin_amdgcn_readlane(e_l, k);  // k is wave-uniform
      const int s = __builtin_amdgcn_readlane(s_l, k);
      const f32x2 xe = __builtin_nontemporal_load(
          (const f32x2*)(edge + (size_t)e * D_FEAT + fo));  // TH=NT
      const f32x2 xs = *(const f32x2*)(node + (size_t)s * D_FEAT + fo);
      ax += fmaxf(xs.x + xe.x, 0.0f);
      ay += fmaxf(xs.y + xe.y, 0.0f);
    }
  }

  const f32x2 xv = *(const f32x2*)(node + (size_t)v * D_FEAT + fo);
  f32x2 r;
  r.x = xv.x + ax;
  r.y = xv.y + ay;
  __builtin_nontemporal_store(r, (f32x2*)(out + (size_t)v * D_FEAT + fo));
}

// ---------------------------------------------------------------------------
// Fallback path (only if d_ws is too small for CSR scratch): init + f32 atomics
// ---------------------------------------------------------------------------
__global__ void k_copy_out(const float* __restrict__ node,
                           float* __restrict__ out, int n4) {
  int i = blockIdx.x * blockDim.x + threadIdx.x;
  const int stride = gridDim.x * blockDim.x;
  for (; i < n4; i += stride)
    ((f32x4*)out)[i] = ((const f32x4*)node)[i];
}

__global__ void k_scatter(const float* __restrict__ node,
                          const float* __restrict__ edge,
                          const int* __restrict__ src,
                          const int* __restrict__ dst,
                          float* __restrict__ out, int E) {
  const long total = (long)E * (D_FEAT / 4);
  long t = (long)blockIdx.x * blockDim.x + threadIdx.x;
  const long stride = (long)gridDim.x * blockDim.x;
  for (; t < total; t += stride) {
    const int e = (int)(t >> 4);
    const int c = ((int)t & 15) * 4;
    const int s = src[e];
    const int d = dst[e];
    const f32x4 xe = __builtin_nontemporal_load(
        (const f32x4*)(edge + (size_t)e * D_FEAT + c));
    const f32x4 xs = *(const f32x4*)(node + (size_t)s * D_FEAT + c);
    float* o = out + (size_t)d * D_FEAT + c;
    __hip_atomic_fetch_add(&o[0], fmaxf(xs.x + xe.x, 0.0f), __ATOMIC_RELAXED,
                           __HIP_MEMORY_SCOPE_AGENT);
    __hip_atomic_fetch_add(&o[1], fmaxf(xs.y + xe.y, 0.0f), __ATOMIC_RELAXED,
                           __HIP_MEMORY_SCOPE_AGENT);
    __hip_atomic_fetch_add(&o[2], fmaxf(xs.z + xe.z, 0.0f), __ATOMIC_RELAXED,
                           __HIP_MEMORY_SCOPE_AGENT);
    __hip_atomic_fetch_add(&o[3], fmaxf(xs.w + xe.w, 0.0f), __ATOMIC_RELAXED,
                           __HIP_MEMORY_SCOPE_AGENT);
  }
}

// ---------------------------------------------------------------------------
extern "C" void kernel_launch(void* const* d_in, const int* in_sizes, int n_in,
                              void* d_out, int out_size, void* d_ws,
                              size_t ws_size, hipStream_t stream) {
  (void)n_in;
  (void)out_size;
  const float* node = (const float*)d_in[0];
  const float* edge = (const float*)d_in[1];
  const int*   src  = (const int*)d_in[2];
  const int*   dst  = (const int*)d_in[3];
  float*       out  = (float*)d_out;

  const int N = in_sizes[0] / D_FEAT;
  const int E = in_sizes[2];

  const size_t need = (size_t)(3 * (size_t)N + 1 + (size_t)E) * sizeof(int);
  if (d_ws != nullptr && ws_size >= need) {
    int* counts  = (int*)d_ws;
    int* offsets = counts + N;         // N+1 entries
    int* cursor  = offsets + (N + 1);  // N entries
    int* edge_of = cursor + N;         // E entries

    k_zero_i32<<<256, 256, 0, stream>>>(counts, N);
    k_hist<<<1024, 256, 0, stream>>>(dst, counts, E);
    k_scan<<<1, 1024, 0, stream>>>(counts, offsets, cursor, N);
    k_fill<<<1024, 256, 0, stream>>>(dst, cursor, edge_of, E);

    const int wavesPerBlock = 256 / WAVE;  // 8 waves per 256-thread block
    const int blocks = (N + wavesPerBlock - 1) / wavesPerBlock;
    k_aggregate<<<blocks, 256, 0, stream>>>(node, edge, src, offsets, edge_of,
                                            out, N);
  } else {
    // Fallback: direct scatter with f32 atomics (higher atomic pressure).
    const int n4 = N * (D_FEAT / 4);
    k_copy_out<<<1024, 256, 0, stream>>>(node, out, n4);
    k_scatter<<<2048, 256, 0, stream>>>(node, edge, src, dst, out, E);
  }
}